// CliffordBatchNorm_49409303773661
// MI455X (gfx1250) — compile-verified
//
#include <hip/hip_runtime.h>
#include <math.h>

typedef __attribute__((ext_vector_type(2))) float v2f;
typedef __attribute__((ext_vector_type(4))) float v4f;
typedef __attribute__((ext_vector_type(8))) float v8f;

#define CH   64      // channels
#define IV   4       // multivector components
#define FEAT 256     // CH*IV floats per spatial position (contiguous)
#define EPSR 1e-5f

// ---------------------------------------------------------------------------
// Pass 1: per-channel sums  S[c][i][j] = sum_p x_i x_j,  sx[c][i] = sum_p x_i
// One wave owns a group of 4 channels (16 "features" = 64 contiguous bytes of
// a position record).  Per iteration the wave covers 8 positions with two
// independent V_WMMA_F32_16X16X4_F32 accumulators (D += A*A^T):
//   lanes 0-15  : features 0-15 of positions p..p+1   (acc0), p+4..p+5 (acc1)
//   lanes 16-31 : features 0-15 of positions p+2..p+3 (acc0), p+6..p+7 (acc1)
// The 4x4 diagonal blocks of D are the per-channel second-moment matrices.
// Block partials are written to d_ws (deterministic; no float atomics).
// ---------------------------------------------------------------------------
__global__ void __launch_bounds__(512) cbn_stats(const float* __restrict__ x,
                                                 float* __restrict__ part,
                                                 int n_oct)
{
    const int tid  = threadIdx.x;
    const int grp  = tid >> 5;   // 16 waves -> 16 channel-groups of 4
    const int lane = tid & 31;
    const int hi   = lane >> 4;
    const int lo   = lane & 15;
    const int off  = grp * 16 + lo;   // feature offset inside a position record

    v8f   acc0 = {};
    v8f   acc1 = {};
    float s    = 0.0f;

    for (int ch = blockIdx.x; ch < n_oct; ch += gridDim.x) {
        const size_t p  = (size_t)ch * 8 + 2 * hi;
        const float  a0 = x[(p + 0) * FEAT + off];
        const float  a1 = x[(p + 1) * FEAT + off];
        const float  b0 = x[(p + 4) * FEAT + off];
        const float  b1 = x[(p + 5) * FEAT + off];
        v2f a; a.x = a0; a.y = a1;
        v2f b; b.x = b0; b.y = b1;
        // D += A * B with A-contents == B-contents  =>  D = sum_p v v^T
        acc0 = __builtin_amdgcn_wmma_f32_16x16x4_f32(
                   false, a, false, a, (short)0, acc0, false, false);
        acc1 = __builtin_amdgcn_wmma_f32_16x16x4_f32(
                   false, b, false, b, (short)0, acc1, false, false);
        s += (a0 + a1) + (b0 + b1);
    }
#pragma unroll
    for (int r = 0; r < 8; ++r) acc0[r] += acc1[r];

    // fold the two position-halves of the first-moment sums
    s += __shfl_xor(s, 16, 32);

    // layout per block: [c:64][20] = { S row-major 16, sumx 4 }
    float*    blk = part + (size_t)blockIdx.x * CH * 20;
    const int cl  = lo >> 2;   // channel within group
    const int j   = lo & 3;

    // C/D layout: acc0[r] = D[r + 8*hi][lo].  Keep only same-channel blocks.
    if ((lo >> 3) == hi) {
        float* dst = blk + (grp * 4 + cl) * 20;
#pragma unroll
        for (int r = 0; r < 8; ++r) {
            const int m = r + 8 * hi;
            if ((m >> 2) == cl) dst[(m & 3) * 4 + j] = acc0[r];
        }
    }
    if (hi == 0) blk[(grp * 4 + cl) * 20 + 16 + j] = s;
}

// ---------------------------------------------------------------------------
// Pass 2: one thread per channel.  Reduce block partials, form covariance,
// 4x4 Cholesky, invert L, compose M = weight[:,:,c] @ Linv and
// shift = bias - M @ mean.   Output: msh[c][20] = { M row-major, shift }.
// ---------------------------------------------------------------------------
__global__ void cbn_solve(const float* __restrict__ part,
                          const float* __restrict__ wgt,   // (I, I, C)
                          const float* __restrict__ bias,  // (I, C)
                          float* __restrict__ msh,
                          int nblk, float inv_n)
{
    const int c = threadIdx.x;
    if (c >= CH) return;

    float S[16], sx[4];
#pragma unroll
    for (int k = 0; k < 16; ++k) S[k] = 0.0f;
#pragma unroll
    for (int i = 0; i < 4; ++i) sx[i] = 0.0f;

    for (int b = 0; b < nblk; ++b) {
        const float* src = part + ((size_t)b * CH + c) * 20;
#pragma unroll
        for (int k = 0; k < 16; ++k) S[k] += src[k];
#pragma unroll
        for (int i = 0; i < 4; ++i) sx[i] += src[16 + i];
    }

    float mean[4];
#pragma unroll
    for (int i = 0; i < 4; ++i) mean[i] = sx[i] * inv_n;

    float cov[4][4];
    for (int i = 0; i < 4; ++i)
        for (int j = 0; j < 4; ++j)
            cov[i][j] = S[i * 4 + j] * inv_n - mean[i] * mean[j];
    for (int i = 0; i < 4; ++i) cov[i][i] += EPSR;

    // Cholesky (lower)
    float L[4][4] = {};
    for (int i = 0; i < 4; ++i)
        for (int j = 0; j <= i; ++j) {
            float sum = cov[i][j];
            for (int k = 0; k < j; ++k) sum -= L[i][k] * L[j][k];
            L[i][j] = (i == j) ? sqrtf(sum) : sum / L[j][j];
        }

    // Linv (lower) by forward substitution on identity
    float Li[4][4] = {};
    for (int j = 0; j < 4; ++j) {
        Li[j][j] = 1.0f / L[j][j];
        for (int i = j + 1; i < 4; ++i) {
            float a = 0.0f;
            for (int k = j; k < i; ++k) a += L[i][k] * Li[k][j];
            Li[i][j] = -a / L[i][i];
        }
    }

    // M = weight[:,:,c] @ Linv ; shift = bias[:,c] - M @ mean
    float M[4][4];
    for (int i = 0; i < 4; ++i)
        for (int k = 0; k < 4; ++k) {
            float a = 0.0f;
            for (int j = 0; j < 4; ++j)
                a += wgt[i * (IV * CH) + j * CH + c] * Li[j][k];
            M[i][k] = a;
        }
    for (int i = 0; i < 4; ++i) {
        float sh = bias[i * CH + c];
        for (int k = 0; k < 4; ++k) sh -= M[i][k] * mean[k];
        msh[c * 20 + 16 + i] = sh;
        for (int k = 0; k < 4; ++k) msh[c * 20 + i * 4 + k] = M[i][k];
    }
}

// ---------------------------------------------------------------------------
// Pass 3: streaming apply.  One lane = one (position, channel): float4 in,
// 16 FMAs, float4 out.  The grid stride is a multiple of 64 so each lane's
// channel is loop-invariant: M/shift are hoisted into 20 registers once.
// x should still be L2-resident from pass 1 (128 MiB < 192 MB L2); output
// uses non-temporal stores so it does not evict x.
// ---------------------------------------------------------------------------
__global__ void __launch_bounds__(256) cbn_apply(const float* __restrict__ x,
                                                 const float* __restrict__ msh,
                                                 float* __restrict__ out,
                                                 int npairs)
{
    __shared__ float lds[CH * 20];
    for (int k = threadIdx.x; k < CH * 20; k += blockDim.x) lds[k] = msh[k];
    __syncthreads();

    const int start  = blockIdx.x * blockDim.x + threadIdx.x;
    const int stride = gridDim.x * blockDim.x;       // multiple of 64

    // channel is invariant along this lane's grid-stride walk
    const float* __restrict__ Msrc = &lds[(start & (CH - 1)) * 20];
    float Mc[20];
#pragma unroll
    for (int k = 0; k < 20; ++k) Mc[k] = Msrc[k];

    const v4f* __restrict__ xin = (const v4f*)x;
    v4f* __restrict__ o4        = (v4f*)out;

    for (int idx = start; idx < npairs; idx += stride) {
        const v4f xv = xin[idx];
        v4f o;
        o.x = Mc[0]  * xv.x + Mc[1]  * xv.y + Mc[2]  * xv.z + Mc[3]  * xv.w + Mc[16];
        o.y = Mc[4]  * xv.x + Mc[5]  * xv.y + Mc[6]  * xv.z + Mc[7]  * xv.w + Mc[17];
        o.z = Mc[8]  * xv.x + Mc[9]  * xv.y + Mc[10] * xv.z + Mc[11] * xv.w + Mc[18];
        o.w = Mc[12] * xv.x + Mc[13] * xv.y + Mc[14] * xv.z + Mc[15] * xv.w + Mc[19];
        __builtin_nontemporal_store(o, &o4[idx]);
    }
}

// ---------------------------------------------------------------------------
extern "C" void kernel_launch(void* const* d_in, const int* in_sizes, int n_in,
                              void* d_out, int out_size, void* d_ws, size_t ws_size,
                              hipStream_t stream)
{
    (void)n_in; (void)out_size;
    const float* x   = (const float*)d_in[0];   // (B,H,W,C,I) fp32
    const float* w   = (const float*)d_in[1];   // (I,I,C)
    const float* b   = (const float*)d_in[2];   // (I,C)
    float*       out = (float*)d_out;

    const int n_pos = in_sizes[0] / FEAT;       // B*H*W = 131072
    const int n_oct = n_pos / 8;                // 8 positions per stats iter

    const size_t per_blk = (size_t)CH * 20 * sizeof(float); // 5 KiB per block
    int nblk = 256;
    if ((size_t)(nblk + 1) * per_blk > ws_size) {
        nblk = (int)(ws_size / per_blk) - 1;
        if (nblk < 1) nblk = 1;
    }
    float* part = (float*)d_ws;                       // [nblk][CH][20]
    float* msh  = part + (size_t)nblk * CH * 20;      // [CH][20]

    cbn_stats<<<nblk, 512, 0, stream>>>(x, part, n_oct);
    cbn_solve<<<1, 64, 0, stream>>>(part, w, b, msh, nblk, 1.0f / (float)n_pos);

    const int npairs = n_pos * CH;
    cbn_apply<<<4096, 256, 0, stream>>>(x, msh, out, npairs);
}